// GCN_2465311228031
// MI455X (gfx1250) — compile-verified
//
#include <hip/hip_runtime.h>

typedef __attribute__((ext_vector_type(2))) float v2f;
typedef __attribute__((ext_vector_type(8))) float v8f;

#define HID 64

// ---------------- degree / normalization ----------------
__global__ void k_init_deg(float* __restrict__ deg, int N) {
    int i = blockIdx.x * blockDim.x + threadIdx.x;
    if (i < N) deg[i] = 1.0f;   // self-loop contributes 1
}

__global__ void k_deg_accum(const int* __restrict__ col, float* __restrict__ deg, int E) {
    int e = blockIdx.x * blockDim.x + threadIdx.x;
    if (e < E) atomicAdd(&deg[col[e]], 1.0f);
}

__global__ void k_rsqrt_inplace(float* __restrict__ deg, int N) {
    int i = blockIdx.x * blockDim.x + threadIdx.x;
    if (i < N) deg[i] = rsqrtf(deg[i]);   // deg >= 1 always
}

// ---------------- GEMM1: x[N,11] @ W1[11,64] -> h[N,64] ----------------
__global__ void k_gemm1(const float* __restrict__ x, const float* __restrict__ W1,
                        float* __restrict__ h, int N) {
    int idx = blockIdx.x * blockDim.x + threadIdx.x;
    if (idx >= N * HID) return;
    int n = idx >> 6;
    int f = idx & (HID - 1);
    float acc = 0.0f;
    #pragma unroll
    for (int k = 0; k < 11; ++k)
        acc = fmaf(x[(size_t)n * 11 + k], W1[(size_t)k * HID + f], acc);
    h[idx] = acc;
}

// ---------------- aggregation ----------------
// agg[n,f] = dis[n]^2 * h[n,f]   (self-loop term; full overwrite)
__global__ void k_selfloop(const float* __restrict__ dis, const float* __restrict__ h,
                           float* __restrict__ agg, int N) {
    int idx = blockIdx.x * blockDim.x + threadIdx.x;
    if (idx >= N * HID) return;
    int n = idx >> 6;
    float d = dis[n];
    agg[idx] = d * d * h[idx];
}

// one thread per (edge, 4-feature chunk): 16 chunks of float4 per edge
__global__ void k_edge_agg(const int* __restrict__ row, const int* __restrict__ col,
                           const float* __restrict__ dis, const float* __restrict__ h,
                           float* __restrict__ agg, int E) {
    long long t = (long long)blockIdx.x * blockDim.x + threadIdx.x;
    if (t >= (long long)E * 16) return;
    int e  = (int)(t >> 4);
    int f4 = (int)(t & 15) << 2;
    int r = row[e];
    int c = col[e];
    float nrm = dis[r] * dis[c];
    const float4 hv = *(const float4*)(h + (size_t)r * HID + f4);
    float* dst = agg + (size_t)c * HID + f4;
    atomicAdd(dst + 0, nrm * hv.x);
    atomicAdd(dst + 1, nrm * hv.y);
    atomicAdd(dst + 2, nrm * hv.z);
    atomicAdd(dst + 3, nrm * hv.w);
}

__global__ void k_bias_relu(float* __restrict__ agg, const float* __restrict__ b, int N) {
    int idx = blockIdx.x * blockDim.x + threadIdx.x;
    if (idx >= N * HID) return;
    int f = idx & (HID - 1);
    float v = agg[idx] + b[f];
    agg[idx] = v > 0.0f ? v : 0.0f;
}

// ---------------- GEMM2 via WMMA: A[N,64] @ B[64,64] -> C[N,64] ----------------
// One wave32 computes one 16x16 output tile with V_WMMA_F32_16X16X4_F32,
// stepping K by 4 (16 wmma ops, fully unrolled).
__global__ __launch_bounds__(256)
void k_gemm_wmma(const float* __restrict__ A, const float* __restrict__ B,
                 float* __restrict__ C, int N) {
    int gtid = blockIdx.x * blockDim.x + threadIdx.x;
    int wave = gtid >> 5;
    int lane = threadIdx.x & 31;
    const int TN = HID / 16;          // 4 column tiles
    int tm = wave / TN;
    int tn = wave % TN;
    if (tm * 16 >= N) return;         // wave-uniform guard (EXEC stays all-ones)

    int l16   = lane & 15;
    int khalf = (lane >> 4) << 1;     // 0 for lanes 0-15, 2 for lanes 16-31
    int m = tm * 16 + l16;            // A-fragment row for this lane
    int n = tn * 16 + l16;            // B/C/D column for this lane

    v8f acc = {0.f, 0.f, 0.f, 0.f, 0.f, 0.f, 0.f, 0.f};

    #pragma unroll
    for (int k = 0; k < HID; k += 4) {
        // A 16x4 fragment: VGPR0/1 hold K={0,1} (lanes 0-15) or K={2,3} (lanes 16-31)
        v2f a = *(const v2f*)(A + (size_t)m * HID + k + khalf);
        // B 4x16 fragment: row-striped across lanes, same K half-split
        v2f b;
        b.x = B[(size_t)(k + khalf + 0) * HID + n];
        b.y = B[(size_t)(k + khalf + 1) * HID + n];
        acc = __builtin_amdgcn_wmma_f32_16x16x4_f32(
            /*neg_a=*/false, a, /*neg_b=*/false, b,
            /*c_mod=*/(short)0, acc, /*reuse_a=*/false, /*reuse_b=*/false);
    }

    // C/D 16x16 f32 layout: VGPR v -> M = v + (lane>=16 ? 8 : 0), N = lane%16
    int mbase = tm * 16 + ((lane >> 4) << 3);
    #pragma unroll
    for (int v = 0; v < 8; ++v)
        C[(size_t)(mbase + v) * HID + n] = acc[v];
}

// ---------------- final: out[n] = h2[n,:] . Wl + bl ----------------
__global__ void k_final(const float* __restrict__ h2, const float* __restrict__ Wl,
                        const float* __restrict__ bl, float* __restrict__ out, int N) {
    int n = blockIdx.x * blockDim.x + threadIdx.x;
    if (n >= N) return;
    const float4* hv = (const float4*)(h2 + (size_t)n * HID);
    const float4* wv = (const float4*)Wl;
    float acc = bl[0];
    #pragma unroll
    for (int q = 0; q < HID / 4; ++q) {
        float4 h = hv[q];
        float4 w = wv[q];
        acc = fmaf(h.x, w.x, acc);
        acc = fmaf(h.y, w.y, acc);
        acc = fmaf(h.z, w.z, acc);
        acc = fmaf(h.w, w.w, acc);
    }
    out[n] = acc;
}

extern "C" void kernel_launch(void* const* d_in, const int* in_sizes, int n_in,
                              void* d_out, int out_size, void* d_ws, size_t ws_size,
                              hipStream_t stream) {
    const float* x  = (const float*)d_in[0];
    const int*   ei = (const int*)  d_in[1];
    const float* W1 = (const float*)d_in[2];
    const float* b1 = (const float*)d_in[3];
    const float* W2 = (const float*)d_in[4];
    const float* b2 = (const float*)d_in[5];
    const float* Wl = (const float*)d_in[6];
    const float* bl = (const float*)d_in[7];

    const int N = in_sizes[0] / 11;
    const int E = in_sizes[1] / 2;
    const int* row = ei;        // edge_index[0] = source
    const int* col = ei + E;    // edge_index[1] = target

    float* out = (float*)d_out;

    // workspace: dis[N] | bufA[N*64] | bufB[N*64]  (~51.6 MB)
    float* dis  = (float*)d_ws;
    float* bufA = dis + N;
    float* bufB = bufA + (size_t)N * HID;

    const int T = 256;
    const int gN   = (N + T - 1) / T;
    const int gE   = (E + T - 1) / T;
    const int gNF  = (N * HID + T - 1) / T;
    const long long eaggT = (long long)E * 16;
    const int gEA  = (int)((eaggT + T - 1) / T);
    const int waves = ((N + 15) / 16) * (HID / 16);          // 25000
    const int gW   = (waves * 32 + T - 1) / T;

    // normalization coefficients
    k_init_deg     <<<gN,  T, 0, stream>>>(dis, N);
    k_deg_accum    <<<gE,  T, 0, stream>>>(col, dis, E);
    k_rsqrt_inplace<<<gN,  T, 0, stream>>>(dis, N);

    // layer 1
    k_gemm1    <<<gNF, T, 0, stream>>>(x, W1, bufA, N);
    k_selfloop <<<gNF, T, 0, stream>>>(dis, bufA, bufB, N);
    k_edge_agg <<<gEA, T, 0, stream>>>(row, col, dis, bufA, bufB, E);
    k_bias_relu<<<gNF, T, 0, stream>>>(bufB, b1, N);

    // layer 2 (dense GEMM on the matrix pipe)
    k_gemm_wmma<<<gW,  T, 0, stream>>>(bufB, W2, bufA, N);
    k_selfloop <<<gNF, T, 0, stream>>>(dis, bufA, bufB, N);
    k_edge_agg <<<gEA, T, 0, stream>>>(row, col, dis, bufA, bufB, E);
    k_bias_relu<<<gNF, T, 0, stream>>>(bufB, b2, N);

    // readout
    k_final<<<gN, T, 0, stream>>>(bufB, Wl, bl, out, N);
}